// NCF_d_12412455485580
// MI455X (gfx1250) — compile-verified
//
#include <hip/hip_runtime.h>

#define BATCH_N   16384
#define MAX_ITEMS 200
#define EMB_DIM   128
#define NTILES    ((MAX_ITEMS + 15) / 16)   // 13

typedef __attribute__((ext_vector_type(2))) float v2f;
typedef __attribute__((ext_vector_type(8))) float v8f;

__global__ __launch_bounds__(128) void ncf_fused_kernel(
    const float* __restrict__ ratings,
    const int*   __restrict__ item_ids,
    const float* __restrict__ noise,
    const float* __restrict__ init_user_emb,
    const float* __restrict__ gmf_emb,
    const float* __restrict__ ncf_emb,
    const float* __restrict__ W0, const float* __restrict__ b0,
    const float* __restrict__ W1, const float* __restrict__ b1,
    const float* __restrict__ W2, const float* __restrict__ b2,
    const float* __restrict__ W3, const float* __restrict__ b3,
    const float* __restrict__ Wn, const float* __restrict__ bn,
    const float* __restrict__ Wf, const float* __restrict__ bf,
    const float* __restrict__ p,
    float* __restrict__ out)
{
    __shared__ float s_row[2 * EMB_DIM];  // [0:128) init_user_emb row, [128:256) noise row
    __shared__ float s_dots[16];          // gmf_user(4) ncf_user(4) gmf_noise(4) ncf_noise(4)
    __shared__ float s_g[4];              // folded GMF head vector
    __shared__ float s_u[8];              // per-row part of hidden pre-activation
    __shared__ int   s_ids[MAX_ITEMS];    // staged item ids for this row

    const int b = blockIdx.x;
    const int t = threadIdx.x;
    const long rowBase = (long)b * MAX_ITEMS;

    // ---- prefetch the ratings row (13 cachelines) while we do per-row work ----
    if (t < 13) __builtin_prefetch(&ratings[rowBase + t * 16], 0, 3);

    // ---- stage input row + item ids in LDS ----
    s_row[t]           = init_user_emb[b * EMB_DIM + t];
    s_row[EMB_DIM + t] = noise[b * EMB_DIM + t];
    for (int i = t; i < MAX_ITEMS; i += 128) s_ids[i] = item_ids[rowBase + i];
    __syncthreads();

    // ---- 16 dot products of length 64: four 64->4 projections ----
    if (t < 16) {
        const int grp = t >> 2;   // 0=gmf_user 1=ncf_user 2=gmf_noise 3=ncf_noise
        const int j   = t & 3;
        const float* W    = (grp == 0) ? W0 : (grp == 1) ? W1 : (grp == 2) ? W2 : W3;
        const float* bias = (grp == 0) ? b0 : (grp == 1) ? b1 : (grp == 2) ? b2 : b3;
        const float* x = &s_row[grp * 64];   // halves laid out in exactly this order
        float acc = bias[j];
        #pragma unroll 8
        for (int k = 0; k < 64; ++k) acc += x[k] * W[j * 64 + k];
        s_dots[t] = acc;
    }
    __syncthreads();

    // g[k] = gmf_user[k]*Wf[k] + gmf_noise[k]*Wf[4+k]
    if (t < 4) s_g[t] = s_dots[t] * Wf[t] + s_dots[8 + t] * Wf[4 + t];
    // u[r] = bn[r] + Wn[r,0:4].ncf_user + Wn[r,4:8].ncf_noise
    if (t < 8) {
        float acc = bn[t];
        #pragma unroll
        for (int j = 0; j < 4; ++j)
            acc += Wn[t * 12 + j] * s_dots[4 + j] + Wn[t * 12 + 4 + j] * s_dots[12 + j];
        s_u[t] = acc;
    }
    __syncthreads();

    // ---- per-wave item loop: 16 items per V_WMMA_F32_16X16X4_F32 ----
    const int wave  = t >> 5;
    const int lane  = t & 31;
    const int row16 = lane & 15;
    const bool hi   = lane >= 16;

    // A matrix = Wn[:,8:12], 8 live rows padded to 16 (loop-invariant).
    // 32-bit A 16x4 layout: VGPR0 = K0 (lanes 0-15) / K2 (lanes 16-31), VGPR1 = K1 / K3.
    v2f A = {0.0f, 0.0f};
    if (row16 < 8) {
        const int kb = 8 + (hi ? 2 : 0);
        A.x = Wn[row16 * 12 + kb];
        A.y = Wn[row16 * 12 + kb + 1];
    }

    const float g0 = s_g[0], g1 = s_g[1], g2 = s_g[2], g3 = s_g[3];
    float u_r[8], wf8[8];
    #pragma unroll
    for (int r = 0; r < 8; ++r) { u_r[r] = s_u[r]; wf8[r] = Wf[8 + r]; }
    const float pv = p[0], bfv = bf[0];

    for (int tile = wave; tile < NTILES; tile += 4) {
        const int item  = tile * 16 + row16;
        const int itemC = (item < MAX_ITEMS) ? item : (MAX_ITEMS - 1);  // keep EXEC full
        const int iid   = s_ids[itemC];                                  // LDS hop, not global

        // Issue all three gathers in one burst.
        // B matrix (4x16 items): low half-wave loads K0,K1; high half loads K2,K3.
        const float2 bv = *(const float2*)&ncf_emb[(long)iid * 4 + (hi ? 2 : 0)];
        const float4 gi = *(const float4*)&gmf_emb[(long)iid * 4];
        float gx = gi.x, gy = gi.y, gz = gi.z, gw = gi.w;
        float rt = ratings[rowBase + itemC];

        // Pin the gather results in VGPRs at full-EXEC scope so the compiler
        // cannot sink these loads into the masked epilogue (one overlapped
        // latency window for all three loads, resolved before the WMMA).
        asm volatile("" : "+v"(gx), "+v"(gy), "+v"(gz), "+v"(gw), "+v"(rt));

        v2f B = {bv.x, bv.y};
        v8f C = {};
        // D[r][item] = sum_k Wn[r][8+k] * ncf_item[item][k]   (exact f32)
        C = __builtin_amdgcn_wmma_f32_16x16x4_f32(
                /*neg_a=*/false, A, /*neg_b=*/false, B,
                /*c_mod=*/(short)0, C, /*reuse_a=*/false, /*reuse_b=*/false);

        float score = bfv + gx * g0 + gy * g1 + gz * g2 + gw * g3;
        #pragma unroll
        for (int r = 0; r < 8; ++r) {
            const float h = u_r[r] + C[r];
            score += fmaxf(h, 0.0f) * wf8[r];
        }
        const float result = pv * rt + score;

        if (!hi && item < MAX_ITEMS)
            out[rowBase + item] = result;
    }
}

extern "C" void kernel_launch(void* const* d_in, const int* in_sizes, int n_in,
                              void* d_out, int out_size, void* d_ws, size_t ws_size,
                              hipStream_t stream) {
    const float* ratings       = (const float*)d_in[0];
    const int*   item_ids      = (const int*)  d_in[1];
    const float* noise         = (const float*)d_in[2];
    const float* init_user_emb = (const float*)d_in[3];
    const float* gmf_emb       = (const float*)d_in[4];
    const float* ncf_emb       = (const float*)d_in[5];
    const float* W0 = (const float*)d_in[6];  const float* b0 = (const float*)d_in[7];
    const float* W1 = (const float*)d_in[8];  const float* b1 = (const float*)d_in[9];
    const float* W2 = (const float*)d_in[10]; const float* b2 = (const float*)d_in[11];
    const float* W3 = (const float*)d_in[12]; const float* b3 = (const float*)d_in[13];
    const float* Wn = (const float*)d_in[14]; const float* bn = (const float*)d_in[15];
    const float* Wf = (const float*)d_in[16]; const float* bf = (const float*)d_in[17];
    const float* p  = (const float*)d_in[18];
    float* out = (float*)d_out;

    ncf_fused_kernel<<<BATCH_N, 128, 0, stream>>>(
        ratings, item_ids, noise, init_user_emb, gmf_emb, ncf_emb,
        W0, b0, W1, b1, W2, b2, W3, b3, Wn, bn, Wf, bf, p, out);
}